// PatchPrototypeLoss_80711025427016
// MI455X (gfx1250) — compile-verified
//
#include <hip/hip_runtime.h>

#define BN 8192      // B*N tokens
#define DIM 768      // feature dim
#define NK 8192      // prototypes

__constant__ const float kTT = 0.07f;   // teacher temp
__constant__ const float kTS = 0.1f;    // student temp

typedef __attribute__((ext_vector_type(2))) float v2f;
typedef __attribute__((ext_vector_type(8))) float v8f;

// ---------------- block reduction helpers (256 threads, wave32) -------------

__device__ __forceinline__ float block_sum256(float x) {
    __shared__ float sh[8];
#pragma unroll
    for (int off = 16; off > 0; off >>= 1) x += __shfl_xor(x, off, 32);
    __syncthreads();                       // protect sh from previous use
    if ((threadIdx.x & 31) == 0) sh[threadIdx.x >> 5] = x;
    __syncthreads();
    float r = sh[0];
#pragma unroll
    for (int i = 1; i < 8; ++i) r += sh[i];
    return r;
}

__device__ __forceinline__ float block_max256(float x) {
    __shared__ float sh[8];
#pragma unroll
    for (int off = 16; off > 0; off >>= 1) x = fmaxf(x, __shfl_xor(x, off, 32));
    __syncthreads();
    if ((threadIdx.x & 31) == 0) sh[threadIdx.x >> 5] = x;
    __syncthreads();
    float r = sh[0];
#pragma unroll
    for (int i = 1; i < 8; ++i) r = fmaxf(r, sh[i]);
    return r;
}

// ---------------- row L2 norms: 1/max(||row||, 1e-12) -----------------------

__global__ void rownorm_kernel(const float* __restrict__ T, const float* __restrict__ S,
                               const float* __restrict__ W,
                               float* __restrict__ tinv, float* __restrict__ sinv,
                               float* __restrict__ winv) {
    const int id  = blockIdx.x;        // 0..3*8192-1
    const int row = id & 8191;
    const int sel = id >> 13;
    const float* src; float* dst;
    if (sel == 0)      { src = T; dst = tinv; }
    else if (sel == 1) { src = S; dst = sinv; }
    else               { src = W; dst = winv; }
    const float* r = src + (size_t)row * DIM;
    float ss = 0.0f;
    for (int i = threadIdx.x; i < DIM; i += 256) { float x = r[i]; ss += x * x; }
    ss = block_sum256(ss);
    if (threadIdx.x == 0) dst[row] = 1.0f / fmaxf(sqrtf(ss), 1e-12f);
}

// ---------------- WMMA 128x128 tile GEMM core (fp32, 16x16x4) ---------------
// C[m0..m0+127][k0..k0+127] = (ainv scaled A rows) . (binv scaled B rows)^T
// 256 threads = 8 waves; wave w owns rows [w*16, w*16+16), all 128 cols.

__device__ __forceinline__ void gemm_tile_core(
    const float* __restrict__ Ag, const float* __restrict__ ainv,
    const float* __restrict__ Bg, const float* __restrict__ binv,
    int m0, int k0, v8f acc[8]) {
    __shared__ float Asl[128][20];   // 16-wide D chunk, padded to 20 (bank-safe)
    __shared__ float Bsl[128][20];
    const int tid  = threadIdx.x;
    const int lane = tid & 31;
    const int wv   = tid >> 5;
    const int lr   = lane & 15;
    const int kp   = (lane >> 4) << 1;     // K-pair select per half-wave
    v8f z = {};
#pragma unroll
    for (int c = 0; c < 8; ++c) acc[c] = z;

    for (int d0 = 0; d0 < DIM; d0 += 16) {
#pragma unroll
        for (int i = 0; i < 2; ++i) {      // 512 float4 loads per matrix / 256 thr
            const int idx = tid + (i << 8);
            const int row = idx >> 2;
            const int cq  = (idx & 3) << 2;
            const float4 va = *reinterpret_cast<const float4*>(
                Ag + (size_t)(m0 + row) * DIM + d0 + cq);
            const float sa = ainv ? ainv[m0 + row] : 1.0f;
            Asl[row][cq + 0] = va.x * sa; Asl[row][cq + 1] = va.y * sa;
            Asl[row][cq + 2] = va.z * sa; Asl[row][cq + 3] = va.w * sa;
            const float4 vb = *reinterpret_cast<const float4*>(
                Bg + (size_t)(k0 + row) * DIM + d0 + cq);
            const float sb = binv ? binv[k0 + row] : 1.0f;
            Bsl[row][cq + 0] = vb.x * sb; Bsl[row][cq + 1] = vb.y * sb;
            Bsl[row][cq + 2] = vb.z * sb; Bsl[row][cq + 3] = vb.w * sb;
        }
        __syncthreads();
#pragma unroll
        for (int kk = 0; kk < 4; ++kk) {
            const int kb = (kk << 2) + kp;
            v2f a;
            a.x = Asl[(wv << 4) + lr][kb];
            a.y = Asl[(wv << 4) + lr][kb + 1];
#pragma unroll
            for (int c = 0; c < 8; ++c) {
                v2f b;
                b.x = Bsl[(c << 4) + lr][kb];
                b.y = Bsl[(c << 4) + lr][kb + 1];
                acc[c] = __builtin_amdgcn_wmma_f32_16x16x4_f32(
                    false, a, false, b, (short)0, acc[c], false, false);
            }
        }
        __syncthreads();
    }
}

// ---------------- teacher GEMM: store logits L, per-block max ---------------

template <typename LT>
__global__ void teacher_gemm_kernel(const float* __restrict__ T, const float* __restrict__ tinv,
                                    const float* __restrict__ W,
                                    LT* __restrict__ L, float* __restrict__ gpart) {
    v8f acc[8];
    const int m0 = blockIdx.y << 7, k0 = blockIdx.x << 7;
    gemm_tile_core(T, tinv, W, nullptr, m0, k0, acc);
    const int lane = threadIdx.x & 31, wv = threadIdx.x >> 5;
    const int lr = lane & 15, hi = lane >> 4;
    float mx = -3.0e38f;
#pragma unroll
    for (int c = 0; c < 8; ++c) {
        const int k = k0 + (c << 4) + lr;
#pragma unroll
        for (int j = 0; j < 8; ++j) {
            const int m = m0 + (wv << 4) + j + (hi << 3);
            const float cv = acc[c][j];
            mx = fmaxf(mx, cv);
            L[(size_t)m * NK + k] = (LT)cv;
        }
    }
    mx = block_max256(mx);
    if (threadIdx.x == 0) gpart[blockIdx.y * gridDim.x + blockIdx.x] = mx;
}

__global__ void reduce_max_kernel(const float* __restrict__ gpart, int n,
                                  float* __restrict__ gmax) {
    float mx = -3.0e38f;
    for (int i = threadIdx.x; i < n; i += 256) mx = fmaxf(mx, gpart[i]);
    mx = block_max256(mx);
    if (threadIdx.x == 0) gmax[0] = mx;
}

__global__ void init_uv_kernel(float* __restrict__ u, float* __restrict__ v) {
    const int i = blockIdx.x * 256 + threadIdx.x;
    u[i] = 1.0f; v[i] = 1.0f;
}

// ---------------- Sinkhorn streaming passes over stored L -------------------
// Q = diag(u) E diag(v), E[k,m] = exp((L[m,k]-gmax)/TT)

template <typename LT>
__global__ void sk_col_kernel(const LT* __restrict__ L, const float* __restrict__ gmaxp,
                              const float* __restrict__ v, float* __restrict__ part) {
    const float g = gmaxp[0];
    const float it = 1.0f / kTT;
    const int k  = blockIdx.x * 256 + threadIdx.x;
    const int mb = blockIdx.y * 512;
    float s = 0.0f;
    for (int m = mb; m < mb + 512; ++m)
        s += __expf(((float)L[(size_t)m * NK + k] - g) * it) * v[m];
    part[(size_t)blockIdx.y * NK + k] = s;
}

__global__ void sk_update_u_kernel(const float* __restrict__ part, float* __restrict__ u) {
    const int k = blockIdx.x * 256 + threadIdx.x;
    float s = 0.0f;
#pragma unroll
    for (int j = 0; j < 16; ++j) s += part[(size_t)j * NK + k];
    const float uk = u[k];
    u[k] = uk / (uk * s + 1e-8f);
}

template <typename LT>
__global__ void sk_row_kernel(const LT* __restrict__ L, const float* __restrict__ gmaxp,
                              const float* __restrict__ u, float* __restrict__ part) {
    const float g = gmaxp[0];
    const float it = 1.0f / kTT;
    const int m  = blockIdx.x;
    const int kb = blockIdx.y * 2048;
    const LT* row = L + (size_t)m * NK;
    float s = 0.0f;
    for (int i = threadIdx.x; i < 2048; i += 256) {
        const int k = kb + i;
        s += __expf(((float)row[k] - g) * it) * u[k];
    }
    s = block_sum256(s);
    if (threadIdx.x == 0) part[(size_t)blockIdx.y * BN + m] = s;
}

__global__ void sk_update_v_kernel(const float* __restrict__ part, float* __restrict__ v) {
    const int m = blockIdx.x * 256 + threadIdx.x;
    float s = 0.0f;
#pragma unroll
    for (int j = 0; j < 4; ++j) s += part[(size_t)j * BN + m];
    const float vm = v[m];
    v[m] = vm / (vm * s + 1e-8f);
}

// ---------------- teacher loss pass: lse, sum u*E*M, sum u*E -----------------

template <typename LT>
__global__ void teacher_final_kernel(const LT* __restrict__ L, const float* __restrict__ gmaxp,
                                     const float* __restrict__ u,
                                     float* __restrict__ pse, float* __restrict__ pwq,
                                     float* __restrict__ psq) {
    const float g = gmaxp[0];
    const float it = 1.0f / kTT;
    const int m  = blockIdx.x;
    const int kb = blockIdx.y * 2048;
    const LT* row = L + (size_t)m * NK;
    float se = 0.0f, wq = 0.0f, sq = 0.0f;
    for (int i = threadIdx.x; i < 2048; i += 256) {
        const int k  = kb + i;
        const float Mv = ((float)row[k] - g) * it;
        const float e  = __expf(Mv);
        const float ue = u[k] * e;
        se += e; wq += ue * Mv; sq += ue;
    }
    se = block_sum256(se);
    wq = block_sum256(wq);
    sq = block_sum256(sq);
    if (threadIdx.x == 0) {
        const size_t o = (size_t)blockIdx.y * BN + m;
        pse[o] = se; pwq[o] = wq; psq[o] = sq;
    }
}

__global__ void teacher_combine_kernel(const float* __restrict__ pse, const float* __restrict__ pwq,
                                       const float* __restrict__ psq, const float* __restrict__ v,
                                       float* __restrict__ sQ, float* __restrict__ tpart) {
    const int m = blockIdx.x * 256 + threadIdx.x;
    float se = 0.0f, wq = 0.0f, sq = 0.0f;
#pragma unroll
    for (int j = 0; j < 4; ++j) {
        const size_t o = (size_t)j * BN + m;
        se += pse[o]; wq += pwq[o]; sq += psq[o];
    }
    sQ[m] = sq;                              // sum_k u_k E[k,m] (final u)
    const float lse = logf(se);
    const float contrib = v[m] * (wq - lse * sq);
    const float s = block_sum256(contrib);
    if (threadIdx.x == 0) tpart[blockIdx.x] = s;
}

// ---------------- student pass: WMMA GEMM fused with streamed Q --------------

template <typename LT>
__global__ void student_pass_kernel(const float* __restrict__ S, const float* __restrict__ sinv,
                                    const float* __restrict__ W,
                                    const LT* __restrict__ L, const float* __restrict__ gmaxp,
                                    const float* __restrict__ u,
                                    float* __restrict__ p1, float* __restrict__ p2) {
    v8f acc[8];
    const int m0 = blockIdx.y << 7, k0 = blockIdx.x << 7;
    gemm_tile_core(S, sinv, W, nullptr, m0, k0, acc);
    const float g   = gmaxp[0];
    const float itt = 1.0f / kTT, its = 1.0f / kTS;
    const int lane = threadIdx.x & 31, wv = threadIdx.x >> 5;
    const int lr = lane & 15, hi = lane >> 4;
    float q1[8], q2[8];
#pragma unroll
    for (int j = 0; j < 8; ++j) { q1[j] = 0.0f; q2[j] = 0.0f; }
#pragma unroll
    for (int c = 0; c < 8; ++c) {
        const int k = k0 + (c << 4) + lr;
        const float uk = u[k];
#pragma unroll
        for (int j = 0; j < 8; ++j) {
            const int m = m0 + (wv << 4) + j + (hi << 3);
            const float Lv = (float)L[(size_t)m * NK + k];
            const float e  = uk * __expf((Lv - g) * itt);
            const float sl = acc[c][j] * its;
            q1[j] += e * sl;             // sum_k u E * (SL/Ts)
            q2[j] += __expf(sl);         // for student row LSE
        }
    }
#pragma unroll
    for (int j = 0; j < 8; ++j) {
#pragma unroll
        for (int off = 1; off < 16; off <<= 1) {
            q1[j] += __shfl_xor(q1[j], off, 32);
            q2[j] += __shfl_xor(q2[j], off, 32);
        }
        if (lr == 0) {
            const int m = m0 + (wv << 4) + j + (hi << 3);
            p1[(size_t)blockIdx.x * BN + m] = q1[j];
            p2[(size_t)blockIdx.x * BN + m] = q2[j];
        }
    }
}

__global__ void student_combine_kernel(const float* __restrict__ p1, const float* __restrict__ p2,
                                       const float* __restrict__ v, const float* __restrict__ sQ,
                                       const int* __restrict__ mask,
                                       float* __restrict__ snum, float* __restrict__ sden) {
    const int m = blockIdx.x * 256 + threadIdx.x;
    float s1 = 0.0f, s2 = 0.0f;
    for (int j = 0; j < 64; ++j) {
        const size_t o = (size_t)j * BN + m;
        s1 += p1[o]; s2 += p2[o];
    }
    const float slse = logf(s2);
    const float pt = -(v[m] * s1 - slse * v[m] * sQ[m]);
    const float mk = (mask[m] != 0) ? 1.0f : 0.0f;
    const float a = block_sum256(pt * mk);
    const float b = block_sum256(mk);
    if (threadIdx.x == 0) { snum[blockIdx.x] = a; sden[blockIdx.x] = b; }
}

// ---------------- KoLeo: Gram GEMM fused with per-row max --------------------

__global__ void koleo_pass_kernel(const float* __restrict__ W, const float* __restrict__ winv,
                                  float* __restrict__ mxp) {
    v8f acc[8];
    const int m0 = blockIdx.y << 7, k0 = blockIdx.x << 7;
    gemm_tile_core(W, winv, W, winv, m0, k0, acc);
    const int lane = threadIdx.x & 31, wv = threadIdx.x >> 5;
    const int lr = lane & 15, hi = lane >> 4;
    float qm[8];
#pragma unroll
    for (int j = 0; j < 8; ++j) qm[j] = -3.0e38f;
#pragma unroll
    for (int c = 0; c < 8; ++c) {
        const int k = k0 + (c << 4) + lr;
#pragma unroll
        for (int j = 0; j < 8; ++j) {
            const int m = m0 + (wv << 4) + j + (hi << 3);
            const float d = (m == k) ? -1.0f : acc[c][j];   // diag excluded
            qm[j] = fmaxf(qm[j], d);
        }
    }
#pragma unroll
    for (int j = 0; j < 8; ++j) {
#pragma unroll
        for (int off = 1; off < 16; off <<= 1)
            qm[j] = fmaxf(qm[j], __shfl_xor(qm[j], off, 32));
        if (lr == 0) {
            const int m = m0 + (wv << 4) + j + (hi << 3);
            mxp[(size_t)blockIdx.x * NK + m] = qm[j];
        }
    }
}

__global__ void koleo_combine_kernel(const float* __restrict__ mxp, float* __restrict__ kpart) {
    const int r = blockIdx.x * 256 + threadIdx.x;
    float mx = -3.0e38f;
    for (int j = 0; j < 64; ++j) mx = fmaxf(mx, mxp[(size_t)j * NK + r]);
    const float d = sqrtf(fmaxf(2.0f - 2.0f * mx, 0.0f));   // unit vectors
    const float val = -logf(d + 1e-8f);
    const float s = block_sum256(val);
    if (threadIdx.x == 0) kpart[blockIdx.x] = s;
}

// ---------------- finalize ---------------------------------------------------

__global__ void finalize_kernel(const float* __restrict__ tpart, const float* __restrict__ snum,
                                const float* __restrict__ sden, const float* __restrict__ kpart,
                                float* __restrict__ out) {
    const int t = threadIdx.x;   // 32 threads = 1 wave
    float a = tpart[t], b = snum[t], c = sden[t], d = kpart[t];
#pragma unroll
    for (int off = 16; off > 0; off >>= 1) {
        a += __shfl_xor(a, off, 32);
        b += __shfl_xor(b, off, 32);
        c += __shfl_xor(c, off, 32);
        d += __shfl_xor(d, off, 32);
    }
    if (t == 0) {
        out[0] = b / fmaxf(c, 1.0f);   // clustering (student) loss
        out[1] = -a / (float)BN;       // teacher proto loss
        out[2] = d / (float)NK;        // koleo
    }
}

// ---------------- host pipeline ---------------------------------------------

template <typename LT>
static void run_pipeline(const float* T, const float* S, const int* mask, const float* W,
                         float* out, LT* L, hipStream_t stream,
                         float* tinv, float* sinv, float* winv, float* u, float* v, float* sQ,
                         float* gpart, float* gmaxv, float* skpart,
                         float* pse, float* pwq, float* psq,
                         float* tpart, float* snum, float* sden, float* kpart,
                         float* p1, float* p2, float* mxp) {
    rownorm_kernel<<<24576, 256, 0, stream>>>(T, S, W, tinv, sinv, winv);
    teacher_gemm_kernel<LT><<<dim3(64, 64), 256, 0, stream>>>(T, tinv, W, L, gpart);
    reduce_max_kernel<<<1, 256, 0, stream>>>(gpart, 4096, gmaxv);
    init_uv_kernel<<<32, 256, 0, stream>>>(u, v);
    for (int it = 0; it < 3; ++it) {
        sk_col_kernel<LT><<<dim3(32, 16), 256, 0, stream>>>(L, gmaxv, v, skpart);
        sk_update_u_kernel<<<32, 256, 0, stream>>>(skpart, u);
        sk_row_kernel<LT><<<dim3(8192, 4), 256, 0, stream>>>(L, gmaxv, u, skpart);
        sk_update_v_kernel<<<32, 256, 0, stream>>>(skpart, v);
    }
    teacher_final_kernel<LT><<<dim3(8192, 4), 256, 0, stream>>>(L, gmaxv, u, pse, pwq, psq);
    teacher_combine_kernel<<<32, 256, 0, stream>>>(pse, pwq, psq, v, sQ, tpart);
    student_pass_kernel<LT><<<dim3(64, 64), 256, 0, stream>>>(S, sinv, W, L, gmaxv, u, p1, p2);
    student_combine_kernel<<<32, 256, 0, stream>>>(p1, p2, v, sQ, mask, snum, sden);
    koleo_pass_kernel<<<dim3(64, 64), 256, 0, stream>>>(W, winv, mxp);
    koleo_combine_kernel<<<32, 256, 0, stream>>>(mxp, kpart);
    finalize_kernel<<<1, 32, 0, stream>>>(tpart, snum, sden, kpart, out);
}

extern "C" void kernel_launch(void* const* d_in, const int* in_sizes, int n_in,
                              void* d_out, int out_size, void* d_ws, size_t ws_size,
                              hipStream_t stream) {
    (void)in_sizes; (void)n_in; (void)out_size;
    const float* T    = (const float*)d_in[0];   // teacher tokens [8,1024,768]
    const float* S    = (const float*)d_in[1];   // student tokens
    const int*   mask = (const int*)d_in[2];     // token masks [8192]
    const float* W    = (const float*)d_in[3];   // prototypes [8192,768]
    float* out = (float*)d_out;

    float* p = (float*)d_ws;
    size_t off = 0;
    auto alloc = [&](size_t n) { float* r = p + off; off += n; return r; };
    float* tinv  = alloc(BN);
    float* sinv  = alloc(BN);
    float* winv  = alloc(NK);
    float* u     = alloc(NK);
    float* v     = alloc(BN);
    float* sQ    = alloc(BN);
    float* gpart = alloc(4096);
    float* gmaxv = alloc(64);
    float* skpart= alloc((size_t)16 * NK);
    float* pse   = alloc((size_t)4 * BN);
    float* pwq   = alloc((size_t)4 * BN);
    float* psq   = alloc((size_t)4 * BN);
    float* tpart = alloc(64);
    float* snum  = alloc(64);
    float* sden  = alloc(64);
    float* kpart = alloc(64);
    float* p1    = alloc((size_t)64 * BN);
    float* p2    = alloc((size_t)64 * BN);
    float* mxp   = alloc((size_t)64 * NK);
    off = (off + 63) & ~(size_t)63;
    const size_t head_bytes = off * sizeof(float);
    const size_t l_elems = (size_t)BN * NK;

    if (ws_size >= head_bytes + l_elems * sizeof(float)) {
        float* L = p + off;
        run_pipeline<float>(T, S, mask, W, out, L, stream,
                            tinv, sinv, winv, u, v, sQ, gpart, gmaxv, skpart,
                            pse, pwq, psq, tpart, snum, sden, kpart, p1, p2, mxp);
    } else {
        // fall back to half-precision logit store (halves workspace + traffic)
        _Float16* L = (_Float16*)(p + off);
        run_pipeline<_Float16>(T, S, mask, W, out, L, stream,
                               tinv, sinv, winv, u, v, sQ, gpart, gmaxv, skpart,
                               pse, pwq, psq, tpart, snum, sden, kpart, p1, p2, mxp);
    }
}